// PV26Criterion_7748121002193
// MI455X (gfx1250) — compile-verified
//
#include <hip/hip_runtime.h>
#include <math.h>

#define BDIM 256
#define NBLK 128            // blocks per 1M-element plane
#define PLANE (1024*1024)
#define NBATCH 8
#define HW 6400             // 80*80
#define NCLS 10
#define NBOX 64
#define EPSV 1e-6f

typedef __attribute__((ext_vector_type(2))) float v2f;
typedef __attribute__((ext_vector_type(8))) float v8f;

// ---------------------------------------------------------------------------
// Wave32 sum via V_WMMA_F32_16X16X4_F32.
// A = 16x4 f32: lane L supplies A[L%16, k] from its 2 VGPRs (k-slot split
// between lane halves).  With B == all-ones and A-VGPR1 == 0:
//   D[m,n] = sum_k A[m,k] = v[m] + v[m+16]   (exact f32 FMA)
// D layout: lanes 0-15 hold rows 0..7 in c[0..7], lanes 16-31 rows 8..15.
// Sum the 8 VGPRs, then one xor-16 shuffle -> full 32-lane sum in all lanes.
// MUST be called with EXEC all-ones (uniform control flow).
// ---------------------------------------------------------------------------
__device__ __forceinline__ float wave_sum_wmma(float v) {
    v2f a; a[0] = v;    a[1] = 0.0f;
    v2f b; b[0] = 1.0f; b[1] = 1.0f;
    v8f c = {};
    c = __builtin_amdgcn_wmma_f32_16x16x4_f32(false, a, false, b,
                                              (short)0, c, false, false);
    float s = c[0] + c[1] + c[2] + c[3] + c[4] + c[5] + c[6] + c[7];
    s += __shfl_xor(s, 16, 32);
    return s;
}

__device__ __forceinline__ float bce_logit(float x, float t) {
    // max(x,0) - x*t + log1p(exp(-|x|))   (numerically stable BCE-with-logits)
    return fmaxf(x, 0.0f) - x * t + log1pf(expf(-fabsf(x)));
}

// ---------------------------------------------------------------------------
// OD: one block per batch image.
// ---------------------------------------------------------------------------
__global__ void od_kernel(const float* __restrict__ det,
                          const float* __restrict__ yolo,
                          float* __restrict__ res) {
    const int b   = blockIdx.x;
    const int tid = threadIdx.x;

    __shared__ int   s_cell[NBOX];
    __shared__ int   s_cls[NBOX];
    __shared__ float s_box[NBOX][4];

    if (tid < NBOX) {
        const float* g = yolo + ((size_t)b * NBOX + tid) * 6;
        const int   cls = (int)g[0];                      // trunc toward zero
        const float cx = fminf(fmaxf(g[1], 0.0f), 1.0f);
        const float cy = fminf(fmaxf(g[2], 0.0f), 1.0f);
        const float w  = fminf(fmaxf(g[3], 0.0f), 1.0f);
        const float h  = fminf(fmaxf(g[4], 0.0f), 1.0f);
        const bool valid = (cls >= 0) && (cls < NCLS) && (g[3] > 0.0f) && (g[4] > 0.0f);
        const int gx = min((int)(cx * 80.0f), 79);
        const int gy = min((int)(cy * 80.0f), 79);
        s_cell[tid] = valid ? (gy * 80 + gx) : -1;
        s_cls[tid]  = cls;
        s_box[tid][0] = cx; s_box[tid][1] = cy; s_box[tid][2] = w; s_box[tid][3] = h;
    }
    __syncthreads();

    const float* base = det + (size_t)b * 15 * HW;
    float obj_s = 0.0f, box_s = 0.0f, cls_s = 0.0f, npos = 0.0f;

    for (int c = tid; c < HW; c += BDIM) {          // 6400/256 = 25, uniform
        int last = -1;                              // last-wins == scatter .set
        #pragma unroll
        for (int j = 0; j < NBOX; ++j)
            if (s_cell[j] == c) last = j;

        const float tobj = (last >= 0) ? 1.0f : 0.0f;
        obj_s += bce_logit(base[4 * HW + c], tobj);

        if (last >= 0) {
            npos += 1.0f;
            #pragma unroll
            for (int k = 0; k < 4; ++k) {
                const float p = 1.0f / (1.0f + expf(-base[k * HW + c]));
                const float d = fabsf(p - s_box[last][k]);
                box_s += (d < 1.0f) ? (0.5f * d * d) : (d - 0.5f);
            }
            float vch[NCLS]; float mx = -3.0e38f;
            #pragma unroll
            for (int j = 0; j < NCLS; ++j) {
                vch[j] = base[(5 + j) * HW + c];
                mx = fmaxf(mx, vch[j]);
            }
            float se = 0.0f;
            #pragma unroll
            for (int j = 0; j < NCLS; ++j) se += expf(vch[j] - mx);
            cls_s += (mx + logf(se)) - vch[s_cls[last]];
        }
    }

    obj_s = wave_sum_wmma(obj_s);
    box_s = wave_sum_wmma(box_s);
    cls_s = wave_sum_wmma(cls_s);
    npos  = wave_sum_wmma(npos);

    __shared__ float red[8][4];
    const int wid = tid >> 5;
    if ((tid & 31) == 0) { red[wid][0] = obj_s; red[wid][1] = box_s; red[wid][2] = cls_s; red[wid][3] = npos; }
    __syncthreads();
    if (tid == 0) {
        float a = 0, bx = 0, cl = 0, np = 0;
        for (int w = 0; w < 8; ++w) { a += red[w][0]; bx += red[w][1]; cl += red[w][2]; np += red[w][3]; }
        float* r = res + b * 4;
        r[0] = a; r[1] = bx; r[2] = cl; r[3] = np;
    }
}

// ---------------------------------------------------------------------------
// DA: grid (NBLK, NBATCH); float4 streaming BCE reduction with ignore=255.
// ---------------------------------------------------------------------------
__global__ void da_partial(const float* __restrict__ lg,
                           const int*  __restrict__ mk,
                           float* __restrict__ part) {
    const int b   = blockIdx.y;
    const int tid = threadIdx.x;
    const float4* l4 = (const float4*)(lg + (size_t)b * PLANE);
    const int4*   m4 = (const int4*)(mk + (size_t)b * PLANE);

    float s = 0.0f, cnt = 0.0f;
    for (int i = blockIdx.x * BDIM + tid; i < PLANE / 4; i += NBLK * BDIM) {
        const float4 x = l4[i];
        const int4   m = m4[i];
        {
            const float v = (m.x != 255) ? 1.0f : 0.0f; const float t = v * (float)(m.x & 1);
            s += bce_logit(x.x, t) * v; cnt += v;
        }
        {
            const float v = (m.y != 255) ? 1.0f : 0.0f; const float t = v * (float)(m.y & 1);
            s += bce_logit(x.y, t) * v; cnt += v;
        }
        {
            const float v = (m.z != 255) ? 1.0f : 0.0f; const float t = v * (float)(m.z & 1);
            s += bce_logit(x.z, t) * v; cnt += v;
        }
        {
            const float v = (m.w != 255) ? 1.0f : 0.0f; const float t = v * (float)(m.w & 1);
            s += bce_logit(x.w, t) * v; cnt += v;
        }
    }
    s   = wave_sum_wmma(s);
    cnt = wave_sum_wmma(cnt);

    __shared__ float red[8][2];
    const int wid = tid >> 5;
    if ((tid & 31) == 0) { red[wid][0] = s; red[wid][1] = cnt; }
    __syncthreads();
    if (tid == 0) {
        float S = 0, C = 0;
        for (int w = 0; w < 8; ++w) { S += red[w][0]; C += red[w][1]; }
        float* p = part + ((size_t)b * NBLK + blockIdx.x) * 2;
        p[0] = S; p[1] = C;
    }
}

// ---------------------------------------------------------------------------
// RM: grid (NBLK, 24 planes); focal + dice partial sums.
// ---------------------------------------------------------------------------
__global__ void rm_partial(const float* __restrict__ lg,
                           const int*  __restrict__ mk,
                           float* __restrict__ part) {
    const int plane = blockIdx.y;          // b*3 + ch, contiguous layout
    const int tid   = threadIdx.x;
    const float4* l4 = (const float4*)(lg + (size_t)plane * PLANE);
    const int4*   m4 = (const int4*)(mk + (size_t)plane * PLANE);

    float focal = 0.0f, inter = 0.0f, psum = 0.0f, tsum = 0.0f, cnt = 0.0f;
    for (int i = blockIdx.x * BDIM + tid; i < PLANE / 4; i += NBLK * BDIM) {
        const float4 x4 = l4[i];
        const int4   m4v = m4[i];
        const float xs[4] = { x4.x, x4.y, x4.z, x4.w };
        const int   ms[4] = { m4v.x, m4v.y, m4v.z, m4v.w };
        #pragma unroll
        for (int k = 0; k < 4; ++k) {
            const float v = (ms[k] != 255) ? 1.0f : 0.0f;
            const float t = v * (float)(ms[k] & 1);
            const float x = xs[k];
            const float bb = bce_logit(x, t);
            const float p  = 1.0f / (1.0f + expf(-x));
            const float pt = p * t + (1.0f - p) * (1.0f - t);
            const float om = 1.0f - pt;
            focal += om * om * bb * v;
            inter += p * t * v;
            psum  += p * v;
            tsum  += t * v;
            cnt   += v;
        }
    }
    focal = wave_sum_wmma(focal);
    inter = wave_sum_wmma(inter);
    psum  = wave_sum_wmma(psum);
    tsum  = wave_sum_wmma(tsum);
    cnt   = wave_sum_wmma(cnt);

    __shared__ float red[8][5];
    const int wid = tid >> 5;
    if ((tid & 31) == 0) {
        red[wid][0] = focal; red[wid][1] = inter; red[wid][2] = psum;
        red[wid][3] = tsum;  red[wid][4] = cnt;
    }
    __syncthreads();
    if (tid == 0) {
        float a0 = 0, a1 = 0, a2 = 0, a3 = 0, a4 = 0;
        for (int w = 0; w < 8; ++w) {
            a0 += red[w][0]; a1 += red[w][1]; a2 += red[w][2]; a3 += red[w][3]; a4 += red[w][4];
        }
        float* p = part + ((size_t)plane * NBLK + blockIdx.x) * 5;
        p[0] = a0; p[1] = a1; p[2] = a2; p[3] = a3; p[4] = a4;
    }
}

// ---------------------------------------------------------------------------
// Final combine: single wave, fully deterministic.
// ---------------------------------------------------------------------------
__global__ void finalize_kernel(const float* __restrict__ daP,
                                const float* __restrict__ rmP,
                                const float* __restrict__ odR,
                                const int* __restrict__ has_det,
                                const int* __restrict__ has_da,
                                const int* __restrict__ has_rm,
                                float* __restrict__ out) {
    __shared__ float da_b[NBATCH], da_w[NBATCH], rm_c[24], rm_w[24];
    const int tid = threadIdx.x;

    if (tid < NBATCH) {
        float S = 0, C = 0;
        const float* p = daP + (size_t)tid * NBLK * 2;
        for (int i = 0; i < NBLK; ++i) { S += p[2 * i]; C += p[2 * i + 1]; }
        da_b[tid] = S / fmaxf(C, 1.0f);
        da_w[tid] = (has_da[tid] != 0 && C > 0.0f) ? 1.0f : 0.0f;
    }
    if (tid < 24) {
        float F = 0, I = 0, P = 0, T = 0, C = 0;
        const float* p = rmP + (size_t)tid * NBLK * 5;
        for (int i = 0; i < NBLK; ++i) {
            F += p[5 * i]; I += p[5 * i + 1]; P += p[5 * i + 2];
            T += p[5 * i + 3]; C += p[5 * i + 4];
        }
        const float rn    = fmaxf(C, 1.0f);
        const float fc    = F / rn;
        const float dice  = 1.0f - (2.0f * I + EPSV) / (P + T + EPSV);
        const float w     = (has_rm[tid] != 0 && C > 0.0f) ? 1.0f : 0.0f;
        rm_c[tid] = (fc + dice) * w;
        rm_w[tid] = w;
    }
    __syncthreads();

    if (tid == 0) {
        float od = 0, ws = 0;
        for (int b = 0; b < NBATCH; ++b) {
            const float obj = odR[b * 4 + 0] / (float)HW;
            const float np  = fmaxf(odR[b * 4 + 3], 1.0f);
            const float box = odR[b * 4 + 1] / (np * 4.0f);
            const float cl  = odR[b * 4 + 2] / np;
            const float w   = (has_det[b] != 0) ? 1.0f : 0.0f;
            od += (obj + box + cl) * w; ws += w;
        }
        od /= fmaxf(ws, 1.0f);

        float da = 0, dw = 0;
        for (int b = 0; b < NBATCH; ++b) { da += da_b[b] * da_w[b]; dw += da_w[b]; }
        da /= fmaxf(dw, 1.0f);

        float rm = 0, rw = 0;
        for (int p = 0; p < 24; ++p) { rm += rm_c[p]; rw += rm_w[p]; }
        rm /= fmaxf(rw, 1.0f);

        const float total = od + da + 2.0f * rm;
        out[0] = total; out[1] = od; out[2] = da; out[3] = rm;
    }
}

// ---------------------------------------------------------------------------
extern "C" void kernel_launch(void* const* d_in, const int* in_sizes, int n_in,
                              void* d_out, int out_size, void* d_ws, size_t ws_size,
                              hipStream_t stream) {
    (void)in_sizes; (void)n_in; (void)out_size; (void)ws_size;
    const float* det  = (const float*)d_in[0];
    const float* yolo = (const float*)d_in[1];
    const float* dal  = (const float*)d_in[2];
    const int*   dam  = (const int*)d_in[3];
    const float* rml  = (const float*)d_in[4];
    const int*   rmm  = (const int*)d_in[5];
    const int*   hdet = (const int*)d_in[6];
    const int*   hda  = (const int*)d_in[7];
    const int*   hrm  = (const int*)d_in[8];
    float* out = (float*)d_out;

    float* ws  = (float*)d_ws;
    float* daP = ws;                                   // 8 * 128 * 2 = 2048 floats
    float* rmP = ws + NBATCH * NBLK * 2;               // 24 * 128 * 5 = 15360 floats
    float* odR = rmP + 24 * NBLK * 5;                  // 8 * 4 = 32 floats

    od_kernel<<<NBATCH, BDIM, 0, stream>>>(det, yolo, odR);
    da_partial<<<dim3(NBLK, NBATCH), BDIM, 0, stream>>>(dal, dam, daP);
    rm_partial<<<dim3(NBLK, 24), BDIM, 0, stream>>>(rml, rmm, rmP);
    finalize_kernel<<<1, 32, 0, stream>>>(daP, rmP, odR, hdet, hda, hrm, out);
}